// GCN_88364657148287
// MI455X (gfx1250) — compile-verified
//
#include <hip/hip_runtime.h>

#define N_NODES 50000
#define N_FEAT  128
#define N_EDGES 600000
#define LDS_STRIDE 132   // 132 % 64 = 4 -> conflict-free A-fragment loads

typedef float v2f __attribute__((ext_vector_type(2)));
typedef float v8f __attribute__((ext_vector_type(8)));

// ---------------------------------------------------------------- zero ws
__global__ void gcn_zero_ws(float* __restrict__ ws, int n) {
    int i = blockIdx.x * blockDim.x + threadIdx.x;
    int stride = gridDim.x * blockDim.x;
    for (; i < n; i += stride) ws[i] = 0.0f;
}

// ------------------------------------------------------------ edge scatter
// One 32-lane group per edge: lane l loads float4 at column 4*l of
// feature[src] (one coalesced 512B row per wave), scales by norm[src],
// atomically accumulates into accum[dst]. Lane 0 counts the in-degree.
__global__ __launch_bounds__(256) void gcn_edge_scatter(
    const float* __restrict__ feature,
    const float* __restrict__ norm,
    const int*   __restrict__ src,
    const int*   __restrict__ dst,
    float* __restrict__ accum,
    float* __restrict__ deg)
{
    long long gt  = (long long)blockIdx.x * blockDim.x + threadIdx.x;
    long long e   = gt >> 5;
    int       sub = (int)(gt & 31);
    if (e >= N_EDGES) return;

    int   s  = src[e];
    int   d  = dst[e];
    float ns = norm[s];

    const float4* fs = (const float4*)(feature + (long long)s * N_FEAT);
    float4 v = fs[sub];

    float* ap = accum + (long long)d * N_FEAT + sub * 4;
    __hip_atomic_fetch_add(ap + 0, v.x * ns, __ATOMIC_RELAXED, __HIP_MEMORY_SCOPE_AGENT);
    __hip_atomic_fetch_add(ap + 1, v.y * ns, __ATOMIC_RELAXED, __HIP_MEMORY_SCOPE_AGENT);
    __hip_atomic_fetch_add(ap + 2, v.z * ns, __ATOMIC_RELAXED, __HIP_MEMORY_SCOPE_AGENT);
    __hip_atomic_fetch_add(ap + 3, v.w * ns, __ATOMIC_RELAXED, __HIP_MEMORY_SCOPE_AGENT);

    if (sub == 0)
        __hip_atomic_fetch_add(deg + d, 1.0f, __ATOMIC_RELAXED, __HIP_MEMORY_SCOPE_AGENT);
}

// -------------------------------------------------- fused node apply + GEMM
// Block = 256 threads = 8 waves; handles nodes [16*blockIdx.x, +16).
// Stage h (16x128) in LDS, then each wave computes one 16x16 output tile
// (j-tile = wave id) via 32x V_WMMA_F32_16X16X4_F32 over K=128.
__global__ __launch_bounds__(256) void gcn_node_gemm(
    const float* __restrict__ feature,
    const float* __restrict__ norm,
    const float* __restrict__ W,      // [128,128], out = h @ W^T
    const float* __restrict__ bias,   // [128]
    const float* __restrict__ accum,
    const float* __restrict__ deg,
    float* __restrict__ out)
{
    __shared__ float hs[16 * LDS_STRIDE];

    const int tid    = threadIdx.x;
    const int m_tile = blockIdx.x;

    // ---- stage h tile: thread t -> row t>>4, 8 contiguous cols
    {
        int row  = tid >> 4;
        int col0 = (tid & 15) * 8;
        long long node = (long long)m_tile * 16 + row;
        float dg = deg[node];
        float nr = norm[node];
        bool  hasdeg = dg > 0.0f;
        const float* sp = hasdeg ? (accum + node * N_FEAT)
                                 : (feature + node * N_FEAT);
        float scale = hasdeg ? nr : 1.0f;
#pragma unroll
        for (int i = 0; i < 8; ++i)
            hs[row * LDS_STRIDE + col0 + i] = sp[col0 + i] * scale;
    }
    __syncthreads();

    const int lane  = tid & 31;
    const int wave  = tid >> 5;        // 0..7 -> column tile
    const int n     = lane & 15;
    const int khalf = lane >> 4;       // 0 or 1
    const int j0    = wave * 16;

    // A: lane holds row m = lane&15, K pair {2*khalf, 2*khalf+1}
    const float* hrow = hs + (lane & 15) * LDS_STRIDE;
    // B[k][j] = W[j0+n][k]  (B = W^T tile), column n = lane&15
    const float* wrow = W + (long long)(j0 + n) * N_FEAT;

    v8f c = {};
#pragma unroll
    for (int k = 0; k < N_FEAT; k += 4) {
        v2f a, b;
        a.x = hrow[k + 2 * khalf];
        a.y = hrow[k + 2 * khalf + 1];
        b.x = wrow[k + 2 * khalf];
        b.y = wrow[k + 2 * khalf + 1];
        c = __builtin_amdgcn_wmma_f32_16x16x4_f32(
                /*neg_a=*/false, a, /*neg_b=*/false, b,
                /*c_mod=*/(short)0, c, /*reuse_a=*/false, /*reuse_b=*/false);
    }

    // C/D layout: VGPR r -> row (r + 8*khalf), col n. Bias + ReLU + store.
    float bv = bias[j0 + n];
#pragma unroll
    for (int r = 0; r < 8; ++r) {
        int m = r + 8 * khalf;
        long long node = (long long)m_tile * 16 + m;
        float v = c[r] + bv;
        out[node * N_FEAT + j0 + n] = v > 0.0f ? v : 0.0f;
    }
}

// ------------------------------------------------------------------ launch
extern "C" void kernel_launch(void* const* d_in, const int* in_sizes, int n_in,
                              void* d_out, int out_size, void* d_ws, size_t ws_size,
                              hipStream_t stream)
{
    (void)in_sizes; (void)n_in; (void)out_size; (void)ws_size;

    const float* feature = (const float*)d_in[0];
    const float* norm    = (const float*)d_in[1];
    const float* W       = (const float*)d_in[2];
    const float* bias    = (const float*)d_in[3];
    const int*   src     = (const int*)d_in[4];
    const int*   dst     = (const int*)d_in[5];
    float*       out     = (float*)d_out;

    float* accum = (float*)d_ws;                       // 50000*128 floats
    float* deg   = accum + (long long)N_NODES * N_FEAT; // 50000 floats

    // 1) zero accum + deg
    int n_zero = N_NODES * N_FEAT + N_NODES;
    gcn_zero_ws<<<1024, 256, 0, stream>>>((float*)d_ws, n_zero);

    // 2) edge scatter: 32 lanes per edge
    long long total_threads = (long long)N_EDGES * 32;
    int blocks = (int)((total_threads + 255) / 256);
    gcn_edge_scatter<<<blocks, 256, 0, stream>>>(feature, norm, src, dst, accum, deg);

    // 3) fused node apply + GEMM (50000 / 16 = 3125 tiles exactly)
    gcn_node_gemm<<<N_NODES / 16, 256, 0, stream>>>(feature, norm, W, bias,
                                                    accum, deg, out);
}